// PlasticSynapse_69466801045994
// MI455X (gfx1250) — compile-verified
//
#include <hip/hip_runtime.h>

// PlasticSynapse fused elementwise update for MI455X (gfx1250).
//
// out[b,o,i] = clip( baseline[o,i]*ALPHA + w[b,o,i]*(1-ALPHA)
//                    + R[b]*|kappa[o,i]|*(post[b,o]*pre[b,i] + SIGMA*noise[b,o,i]),
//                    0, 1 )
//
// Pure streaming kernel: ~776 MB of once-touched traffic -> ~33 us roofline
// at 23.3 TB/s HBM. No WMMA on purpose: the |kappa| elementwise mask sits
// between the rank-1 product and the accumulate, which D = A*B + C cannot
// fuse, so matrix ops would add work without saving a byte of HBM traffic.
// The gfx1250 features that matter: b128 vector memory ops, non-temporal
// cache hints (th:TH_LOAD_NT / th:TH_STORE_NT) on the streams so they do not
// evict the reused pre/post/R lines, and global_prefetch_b8.

typedef float v4f __attribute__((ext_vector_type(4)));

namespace {
constexpr int   B_DIM       = 64;
constexpr int   OUT_DIM     = 1024;
constexpr int   IN_DIM      = 1024;
constexpr int   BGROUPS     = 4;                 // batch split for occupancy
constexpr int   B_PER_GROUP = B_DIM / BGROUPS;   // 16 batches per block
constexpr int   PF_DIST     = 3;                 // prefetch distance (batches)
constexpr float ALPHA       = 0.01f;             // DT_W / TAU_W
constexpr float ONE_M_ALPHA = 1.0f - 0.01f;
constexpr float SIGMA       = 0.14142135623730953f; // sqrt(2/ALPHA) * 0.01
} // namespace

__global__ __launch_bounds__(256)
void plastic_synapse_fused(const float* __restrict__ w,
                           const float* __restrict__ baseline,
                           const float* __restrict__ Rg,
                           const float* __restrict__ pre,
                           const float* __restrict__ post,
                           const float* __restrict__ kappa,
                           const float* __restrict__ noise,
                           float* __restrict__ out)
{
    const int o  = blockIdx.x;             // one OUT row per block
    const int i0 = threadIdx.x << 2;       // 4 consecutive IN elements per lane
    const int p  = o * IN_DIM + i0;        // offset within the (OUT,IN) plane
    const int b0 = blockIdx.y * B_PER_GROUP;
    const int plane = OUT_DIM * IN_DIM;    // 1M elements
    const int pbase = b0 * plane + p;      // max ~64M elements, fits i32

    // baseline/kappa are touched once per (o,i) in this block's batch range:
    // fold the constants and hold them in registers across the batch loop.
    v4f base4 = *reinterpret_cast<const v4f*>(baseline + p);
    v4f kap4  = *reinterpret_cast<const v4f*>(kappa + p);
    v4f basea = base4 * ALPHA;
    v4f akap;
    akap.x = __builtin_fabsf(kap4.x);
    akap.y = __builtin_fabsf(kap4.y);
    akap.z = __builtin_fabsf(kap4.z);
    akap.w = __builtin_fabsf(kap4.w);

    // Block-uniform scalars: hoist all of them so the inner loop carries no
    // s_load / s_wait_kmcnt. Fully-unrolled locals promote to SGPRs.
    float sR[B_PER_GROUP], sPost[B_PER_GROUP];
#pragma unroll
    for (int k = 0; k < B_PER_GROUP; ++k) {
        sR[k]    = Rg[b0 + k];
        sPost[k] = post[(b0 + k) * OUT_DIM + o];
    }

    // Fully unrolled batch loop: compile-time prefetch guard (no branches),
    // many NT b128 loads in flight, adjacent bodies 4 MB apart (fits the
    // signed 24-bit instruction offset).
#pragma unroll
    for (int k = 0; k < B_PER_GROUP; ++k) {
        const int off = pbase + k * plane;

        // Pull a future batch's cachelines toward the WGP while this
        // iteration's loads are in flight (gfx1250 global_prefetch_b8).
        // Guard is compile-time: never prefetch past the allocation
        // (WGP-scope prefetches are non-speculative per the ISA).
        if (k + PF_DIST < B_PER_GROUP) {
            __builtin_prefetch(w     + off + PF_DIST * plane, 0, 1);
            __builtin_prefetch(noise + off + PF_DIST * plane, 0, 1);
        }

        // Once-touched streams: non-temporal so ~768 MB of traffic does not
        // evict the reused pre/post/R lines from WGP$/L2.
        v4f w4 = __builtin_nontemporal_load(reinterpret_cast<const v4f*>(w + off));
        v4f n4 = __builtin_nontemporal_load(reinterpret_cast<const v4f*>(noise + off));

        // Reused, small: regular (temporal) load, L2-resident across blocks.
        v4f pre4 = *reinterpret_cast<const v4f*>(pre + (b0 + k) * IN_DIM + i0);

        const v4f coef = akap * sR[k];     // R[b] * |kappa|
        const v4f hebb = pre4 * sPost[k];  // post[b,o] * pre[b,i]

        v4f res;
        res.x = __builtin_fmaf(coef.x, __builtin_fmaf(SIGMA, n4.x, hebb.x),
                               __builtin_fmaf(w4.x, ONE_M_ALPHA, basea.x));
        res.y = __builtin_fmaf(coef.y, __builtin_fmaf(SIGMA, n4.y, hebb.y),
                               __builtin_fmaf(w4.y, ONE_M_ALPHA, basea.y));
        res.z = __builtin_fmaf(coef.z, __builtin_fmaf(SIGMA, n4.z, hebb.z),
                               __builtin_fmaf(w4.z, ONE_M_ALPHA, basea.z));
        res.w = __builtin_fmaf(coef.w, __builtin_fmaf(SIGMA, n4.w, hebb.w),
                               __builtin_fmaf(w4.w, ONE_M_ALPHA, basea.w));

        // clip to [0, WEIGHT_BOUND] in one VALU op per component (v_med3)
        res.x = __builtin_amdgcn_fmed3f(res.x, 0.0f, 1.0f);
        res.y = __builtin_amdgcn_fmed3f(res.y, 0.0f, 1.0f);
        res.z = __builtin_amdgcn_fmed3f(res.z, 0.0f, 1.0f);
        res.w = __builtin_amdgcn_fmed3f(res.w, 0.0f, 1.0f);

        __builtin_nontemporal_store(res, reinterpret_cast<v4f*>(out + off));
    }
}

extern "C" void kernel_launch(void* const* d_in, const int* in_sizes, int n_in,
                              void* d_out, int out_size, void* d_ws, size_t ws_size,
                              hipStream_t stream) {
    // setup_inputs() order: w, baseline, R, pre, post, kappa, noise (all f32)
    const float* w        = (const float*)d_in[0];
    const float* baseline = (const float*)d_in[1];
    const float* R        = (const float*)d_in[2];
    const float* pre      = (const float*)d_in[3];
    const float* post     = (const float*)d_in[4];
    const float* kappa    = (const float*)d_in[5];
    const float* noise    = (const float*)d_in[6];
    float* out            = (float*)d_out;

    (void)in_sizes; (void)n_in; (void)out_size; (void)d_ws; (void)ws_size;

    dim3 grid(OUT_DIM, BGROUPS);   // 1024 x 4 = 4096 workgroups
    dim3 block(IN_DIM / 4);        // 256 threads = 8 wave32

    hipLaunchKernelGGL(plastic_synapse_fused, grid, block, 0, stream,
                       w, baseline, R, pre, post, kappa, noise, out);
}